// BetaGNN_16844861734926
// MI455X (gfx1250) — compile-verified
//
#include <hip/hip_runtime.h>

typedef __attribute__((ext_vector_type(2))) float v2f;
typedef __attribute__((ext_vector_type(8))) float v8f;

#define HID 128

// ---------------------------------------------------------------- zero fill
__global__ __launch_bounds__(256) void zero_f4(float4* __restrict__ p, int n4) {
    int i = blockIdx.x * 256 + threadIdx.x;
    if (i < n4) p[i] = make_float4(0.f, 0.f, 0.f, 0.f);
}

// --------------------------------------------- H = relu(X @ W_in + b_in)
// block = 256 threads = 8 waves; block covers 16 rows x 128 cols.
// wave w computes the 16x16 tile at cols [16w, 16w+16) via 32 chained
// v_wmma_f32_16x16x4_f32 ops over K=128.
__global__ __launch_bounds__(256) void gemm_bias_relu(
    const float* __restrict__ X, const float* __restrict__ W,
    const float* __restrict__ bias, float* __restrict__ H) {
    const int row0 = blockIdx.x * 16;
    const int lane = threadIdx.x & 31;
    const int wave = threadIdx.x >> 5;
    const int col0 = wave * 16;
    const int m    = lane & 15;     // A-row / B-col / D-col within tile
    const int half = lane >> 4;     // K / M half selector

    const float* arow = X + (size_t)(row0 + m) * HID;
    v8f acc = {};
    for (int k0 = 0; k0 < HID; k0 += 4) {
        const int ka = k0 + half * 2;
        v2f a; a.x = arow[ka];                        a.y = arow[ka + 1];
        v2f b; b.x = W[(size_t)ka * HID + col0 + m];  b.y = W[(size_t)(ka + 1) * HID + col0 + m];
        acc = __builtin_amdgcn_wmma_f32_16x16x4_f32(false, a, false, b,
                                                    (short)0, acc, false, false);
    }
    const float bz = bias[col0 + m];
#pragma unroll
    for (int i = 0; i < 8; ++i) {
        float v = acc[i] + bz;
        v = v > 0.f ? v : 0.f;
        H[(size_t)(row0 + half * 8 + i) * HID + col0 + m] = v;
    }
}

// ----------------------------------- Hout[r] += vals[e] * Hin[col[e]]
// One wave per edge: lane holds 4 consecutive floats (float4 gather,
// 4x global_atomic_add_f32 scatter). H tables are L2-resident (51MB << 192MB).
__global__ __launch_bounds__(256) void spmm_scatter(
    const int* __restrict__ row, const int* __restrict__ col,
    const float* __restrict__ vals, const float* __restrict__ Hin,
    float* __restrict__ Hout, int E) {
    const int e = blockIdx.x * 8 + (threadIdx.x >> 5);
    if (e >= E) return;
    const int lane = threadIdx.x & 31;
    const int r = row[e];
    const int c = col[e];
    const float v = vals[e];
    const float4 h = reinterpret_cast<const float4*>(Hin + (size_t)c * HID)[lane];
    float* dst = Hout + (size_t)r * HID + lane * 4;
    atomicAdd(dst + 0, v * h.x);
    atomicAdd(dst + 1, v * h.y);
    atomicAdd(dst + 2, v * h.z);
    atomicAdd(dst + 3, v * h.w);
}

// ------- out = relu(AH@W1 + A2H@W2) @ W_out + b_out, fused per 16-row block
__global__ __launch_bounds__(256) void mp_relu_out(
    const float* __restrict__ AH, const float* __restrict__ A2H,
    const float* __restrict__ W1, const float* __restrict__ W2,
    const float* __restrict__ Wout, const float* __restrict__ bout,
    float* __restrict__ out) {
    __shared__ float partial[16];
    const int row0 = blockIdx.x * 16;
    const int lane = threadIdx.x & 31;
    const int wave = threadIdx.x >> 5;
    const int col0 = wave * 16;
    const int m    = lane & 15;
    const int half = lane >> 4;

    if (threadIdx.x < 16) partial[threadIdx.x] = 0.f;
    __syncthreads();

    const float* a1row = AH  + (size_t)(row0 + m) * HID;
    const float* a2row = A2H + (size_t)(row0 + m) * HID;
    v8f acc = {};
    for (int k0 = 0; k0 < HID; k0 += 4) {
        const int ka = k0 + half * 2;
        v2f a, b;
        a.x = a1row[ka];                        a.y = a1row[ka + 1];
        b.x = W1[(size_t)ka * HID + col0 + m];  b.y = W1[(size_t)(ka + 1) * HID + col0 + m];
        acc = __builtin_amdgcn_wmma_f32_16x16x4_f32(false, a, false, b,
                                                    (short)0, acc, false, false);
        a.x = a2row[ka];                        a.y = a2row[ka + 1];
        b.x = W2[(size_t)ka * HID + col0 + m];  b.y = W2[(size_t)(ka + 1) * HID + col0 + m];
        acc = __builtin_amdgcn_wmma_f32_16x16x4_f32(false, a, false, b,
                                                    (short)0, acc, false, false);
    }

    const float w = Wout[col0 + m];
#pragma unroll
    for (int i = 0; i < 8; ++i) {
        float h2 = acc[i] > 0.f ? acc[i] : 0.f;
        float p = h2 * w;
        p += __shfl_xor(p, 1, 32);   // reduce across the 16-lane half that
        p += __shfl_xor(p, 2, 32);   // shares this output row
        p += __shfl_xor(p, 4, 32);
        p += __shfl_xor(p, 8, 32);
        if (m == 0) atomicAdd(&partial[half * 8 + i], p);   // ds_add_f32
    }
    __syncthreads();
    if (threadIdx.x < 16)
        out[row0 + threadIdx.x] = partial[threadIdx.x] + bout[0];
}

// ---------------------------------------------------------------- driver
extern "C" void kernel_launch(void* const* d_in, const int* in_sizes, int n_in,
                              void* d_out, int out_size, void* d_ws, size_t ws_size,
                              hipStream_t stream) {
    (void)n_in; (void)out_size; (void)ws_size;
    const float* X    = (const float*)d_in[0];
    const int*   row  = (const int*)  d_in[1];
    const int*   col  = (const int*)  d_in[2];
    const float* vals = (const float*)d_in[3];
    const float* W_in = (const float*)d_in[4];
    const float* b_in = (const float*)d_in[5];
    const float* W1   = (const float*)d_in[6];
    const float* W2   = (const float*)d_in[7];
    const float* Wout = (const float*)d_in[8];
    const float* bout = (const float*)d_in[9];
    float* out = (float*)d_out;

    const int N = in_sizes[0] / HID;     // 100000
    const int E = in_sizes[1];           // 1600000
    const size_t nodeElems = (size_t)N * HID;

    float* buf0 = (float*)d_ws;          // H, later reused as A2H
    float* buf1 = buf0 + nodeElems;      // AH

    const int n4     = (int)(nodeElems / 4);
    const int zgrid  = (n4 + 255) / 256;
    const int mtiles = N / 16;           // 6250 (N % 16 == 0)
    const int egrid  = (E + 7) / 8;      // 8 edges (waves) per block

    zero_f4       <<<zgrid,  256, 0, stream>>>((float4*)buf1, n4);           // AH = 0
    gemm_bias_relu<<<mtiles, 256, 0, stream>>>(X, W_in, b_in, buf0);         // H
    spmm_scatter  <<<egrid,  256, 0, stream>>>(row, col, vals, buf0, buf1, E); // AH
    zero_f4       <<<zgrid,  256, 0, stream>>>((float4*)buf0, n4);           // A2H = 0
    spmm_scatter  <<<egrid,  256, 0, stream>>>(row, col, vals, buf1, buf0, E); // A2H
    mp_relu_out   <<<mtiles, 256, 0, stream>>>(buf1, buf0, W1, W2, Wout, bout, out);
}